// FlowEmbedder_67851893342237
// MI455X (gfx1250) — compile-verified
//
#include <hip/hip_runtime.h>
#include <math.h>

typedef __attribute__((ext_vector_type(2))) float v2f;
typedef __attribute__((ext_vector_type(8))) float v8f;

#define STEP_SIZE     0.1f
#define FLOW_STRENGTH 5.0f
#define NNODES        1024
#define LROW          17          // padded LDS row stride (words) -> conflict-free
#define WAVES_PER_BLK 8
#define PAIRS_PER_WAVE 16
#define LOG2E         1.4426950408889634f

// gfx1250 hardware tanh (confirmed lowering to v_tanh_f32)
__device__ __forceinline__ float fast_tanh(float x) {
#if __has_builtin(__builtin_amdgcn_tanhf)
    return __builtin_amdgcn_tanhf(x);
#else
    const float e = __builtin_amdgcn_exp2f(x * (2.0f * LOG2E));
    return 1.0f - 2.0f * __builtin_amdgcn_rcpf(e + 1.0f);
#endif
}

__global__ __launch_bounds__(256) void FlowEmbedder_kernel(
    const float* __restrict__ pts,
    const float* __restrict__ W1, const float* __restrict__ b1,
    const float* __restrict__ W2, const float* __restrict__ b2,
    const float* __restrict__ W3, const float* __restrict__ b3,
    float* __restrict__ out)
{
    // zero-padded, B-layout-ready weights (shared by all waves in block)
    __shared__ float sB1[4 * 16];     // layer1 B: 4x16 (K x N), rows 2,3 = 0
    __shared__ float sB2[16 * 16];    // layer2 B: 16x16, real 10x10
    __shared__ float sB3[16 * 16];    // layer3 B: 16x16, real 10x2
    __shared__ float sb1[16], sb2[16], sb3[16];
    __shared__ float sH[WAVES_PER_BLK][16 * LROW];   // per-wave relayout scratch

    const int tid   = threadIdx.x;
    const int lane  = tid & 31;
    const int wave  = tid >> 5;
    const int m     = lane & 15;   // A-row / step index
    const int n     = lane & 15;   // B / C/D column index
    const int upper = lane >> 4;   // 0 = K low half, 1 = K high half
    const int k0    = 2 * upper;   // per-lane K base within a K=4 chunk

    // ---------------- one-time block staging (branch-free fills) ----------------
    {
        const int kk = tid >> 4, cc = tid & 15;
        if (tid < 64)
            sB1[tid] = (kk < 2 && cc < 10) ? W1[kk * 10 + cc] : 0.f;
        sB2[tid] = (kk < 10 && cc < 10) ? W2[kk * 10 + cc] : 0.f;
        sB3[tid] = (kk < 10 && cc < 2)  ? W3[kk * 2  + cc] : 0.f;
        if (tid < 16) {
            sb1[tid] = (tid < 10) ? b1[tid] : 0.f;
            sb2[tid] = (tid < 10) ? b2[tid] : 0.f;
            sb3[tid] = (tid < 2)  ? b3[tid] : 0.f;
        }
    }
    __syncthreads();

    // ---------------- per-wave hoisted operands (loop-invariant) ----------------
    v2f B1r, B2r[3], B3r[3];
    B1r.x = sB1[(k0 + 0) * 16 + n];
    B1r.y = sB1[(k0 + 1) * 16 + n];
    #pragma unroll
    for (int c = 0; c < 3; ++c) {
        const int kc = 4 * c + k0;
        B2r[c].x = sB2[(kc + 0) * 16 + n];
        B2r[c].y = sB2[(kc + 1) * 16 + n];
        B3r[c].x = sB3[(kc + 0) * 16 + n];
        B3r[c].y = sB3[(kc + 1) * 16 + n];
    }
    const float bias1 = sb1[n], bias2 = sb2[n], bias3 = sb3[n];

    float* H = sH[wave];
    const float2* __restrict__ pts2 = (const float2*)pts;

    const int base = (blockIdx.x * WAVES_PER_BLK + wave) * PAIRS_PER_WAVE;
    float res = 0.f;

    for (int p = 0; p < PAIRS_PER_WAVE; ++p) {
        // wave-uniform pair id -> scalar loads of the two points
        const int swid = __builtin_amdgcn_readfirstlane(base + p);
        const int i = swid >> 10;
        const int j = swid & (NNODES - 1);
        const float2 Pi = pts2[i];
        const float2 Pj = pts2[j];

        const float dx  = Pj.x - Pi.x, dy = Pj.y - Pi.y;
        const float euc = __builtin_amdgcn_sqrtf(dx * dx + dy * dy);
        const float se  = (euc > 0.f) ? euc : 1.f;
        const float rse = __builtin_amdgcn_rcpf(se);
        const float ux  = dx * rse, uy = dy * rse;

        // ---- Layer 1: steps(16x2, Kpad=4) @ W1 + b1 ----
        v2f a;
        {
            const float t = (float)m * STEP_SIZE;
            a.x = upper ? 0.f : fmaf(t, ux, Pi.x);   // K=0 | K=2 pad
            a.y = upper ? 0.f : fmaf(t, uy, Pi.y);   // K=1 | K=3 pad
        }
        v8f acc;
        #pragma unroll
        for (int q = 0; q < 8; ++q) acc[q] = bias1;
        acc = __builtin_amdgcn_wmma_f32_16x16x4_f32(false, a, false, B1r, (short)0, acc, false, false);

        #pragma unroll
        for (int q = 0; q < 8; ++q)
            H[(upper * 8 + q) * LROW + n] = fast_tanh(acc[q]);

        // ---- Layer 2: h1 @ W2 + b2 (3 chained K=4 WMMAs) ----
        #pragma unroll
        for (int q = 0; q < 8; ++q) acc[q] = bias2;
        #pragma unroll
        for (int c = 0; c < 3; ++c) {
            v2f av;
            av.x = H[m * LROW + 4 * c + k0];
            av.y = H[m * LROW + 4 * c + k0 + 1];
            acc = __builtin_amdgcn_wmma_f32_16x16x4_f32(false, av, false, B2r[c], (short)0, acc, false, false);
        }
        #pragma unroll
        for (int q = 0; q < 8; ++q)
            H[(upper * 8 + q) * LROW + n] = fast_tanh(acc[q]);   // same-wave DS in-order

        // ---- Layer 3: h2 @ W3 + b3 ----
        #pragma unroll
        for (int q = 0; q < 8; ++q) acc[q] = bias3;
        #pragma unroll
        for (int c = 0; c < 3; ++c) {
            v2f av;
            av.x = H[m * LROW + 4 * c + k0];
            av.y = H[m * LROW + 4 * c + k0 + 1];
            acc = __builtin_amdgcn_wmma_f32_16x16x4_f32(false, av, false, B3r[c], (short)0, acc, false, false);
        }
        #pragma unroll
        for (int q = 0; q < 8; ++q)                  // unconditional: no exec branch
            H[(upper * 8 + q) * LROW + n] = acc[q];

        // ---- per-step cost (all 32 lanes; halves duplicate harmlessly) ----
        const float fx   = H[m * LROW + 0];
        const float fy   = H[m * LROW + 1];
        const float dot  = fx * dx + fy * dy;
        const float einv = __builtin_amdgcn_exp2f(-dot * LOG2E);   // exp(-dot)
        // per_step = (S/F) * (1 + (F-1)*exp(-dot))  -> division-free
        const float per  = (STEP_SIZE / FLOW_STRENGTH) *
                           fmaf(FLOW_STRENGTH - 1.f, einv, 1.f);
        const float nvalid = floorf(euc * (1.0f / STEP_SIZE));
        float cost = ((float)m <= nvalid) ? per : 0.f;

        #pragma unroll
        for (int off = 1; off < 16; off <<= 1)
            cost += __shfl_xor(cost, off, 32);       // each 16-half holds the sum

        const float total = (euc > 0.f) ? cost : 0.f;
        res = (lane == p) ? total : res;             // collect, store coalesced later
    }

    if (lane < PAIRS_PER_WAVE)
        out[base + lane] = res;                      // one coalesced 64B store/wave
}

extern "C" void kernel_launch(void* const* d_in, const int* in_sizes, int n_in,
                              void* d_out, int out_size, void* d_ws, size_t ws_size,
                              hipStream_t stream) {
    (void)in_sizes; (void)n_in; (void)out_size; (void)d_ws; (void)ws_size;
    const float* pts = (const float*)d_in[0];
    const float* W1  = (const float*)d_in[1];
    const float* b1  = (const float*)d_in[2];
    const float* W2  = (const float*)d_in[3];
    const float* b2  = (const float*)d_in[4];
    const float* W3  = (const float*)d_in[5];
    const float* b3  = (const float*)d_in[6];
    float* out = (float*)d_out;

    const int total_pairs = NNODES * NNODES;
    const int pairs_per_block = WAVES_PER_BLK * PAIRS_PER_WAVE;   // 128
    const int blocks = total_pairs / pairs_per_block;             // 8192
    FlowEmbedder_kernel<<<dim3(blocks), dim3(256), 0, stream>>>(
        pts, W1, b1, W2, b2, W3, b3, out);
}